// Megnet_GlobalModel_74818330296972
// MI455X (gfx1250) — compile-verified
//
#include <hip/hip_runtime.h>
#include <hip/hip_bf16.h>

typedef float v2f __attribute__((ext_vector_type(2)));
typedef float v8f __attribute__((ext_vector_type(8)));

#define DIM   256
#define DIM3  768

// ---------------------------------------------------------------------------
// 1) edge -> node scatter-add (atomics land in L2: node_sum is 102MB < 192MB L2)
//    4 edges per 256-thread block, 64 threads per edge, float4 per thread.
// ---------------------------------------------------------------------------
__global__ void __launch_bounds__(256)
edge_scatter(const float* __restrict__ edge_attr, const int* __restrict__ edge_src,
             float* __restrict__ node_sum, float* __restrict__ node_cnt, int n_edges) {
  int e = blockIdx.x * 4 + (threadIdx.x >> 6);
  if (e >= n_edges) return;
  int j = threadIdx.x & 63;
  int src = edge_src[e];
  const float4 v = *(const float4*)(edge_attr + (long)e * DIM + j * 4);
  float* dst = node_sum + (long)src * DIM + j * 4;
  atomicAdd(dst + 0, v.x);
  atomicAdd(dst + 1, v.y);
  atomicAdd(dst + 2, v.z);
  atomicAdd(dst + 3, v.w);
  if (j == 0) atomicAdd(node_cnt + src, 1.0f);
}

// ---------------------------------------------------------------------------
// 2) node -> graph scatter: mean(node_sum) and x into per-graph accumulators
// ---------------------------------------------------------------------------
__global__ void __launch_bounds__(256)
node_scatter(const float* __restrict__ x, const int* __restrict__ batch,
             const float* __restrict__ node_sum, const float* __restrict__ node_cnt,
             float* __restrict__ gsum_e, float* __restrict__ gsum_v,
             float* __restrict__ gcnt, int n_nodes) {
  int n = blockIdx.x * 4 + (threadIdx.x >> 6);
  if (n >= n_nodes) return;
  int j = threadIdx.x & 63;
  int g = batch[n];
  float inv = 1.0f / fmaxf(node_cnt[n], 1.0f);
  const float4 s = *(const float4*)(node_sum + (long)n * DIM + j * 4);
  float* de = gsum_e + (long)g * DIM + j * 4;
  atomicAdd(de + 0, s.x * inv);
  atomicAdd(de + 1, s.y * inv);
  atomicAdd(de + 2, s.z * inv);
  atomicAdd(de + 3, s.w * inv);
  const float4 xv = *(const float4*)(x + (long)n * DIM + j * 4);
  float* dv = gsum_v + (long)g * DIM + j * 4;
  atomicAdd(dv + 0, xv.x);
  atomicAdd(dv + 1, xv.y);
  atomicAdd(dv + 2, xv.z);
  atomicAdd(dv + 3, xv.w);
  if (j == 0) atomicAdd(gcnt + g, 1.0f);
}

// ---------------------------------------------------------------------------
// 3) comb = concat(gsum_e/cnt, gsum_v/cnt, u)   [G x 768]
// ---------------------------------------------------------------------------
__global__ void __launch_bounds__(256)
comb_build(const float* __restrict__ gsum_e, const float* __restrict__ gsum_v,
           const float* __restrict__ gcnt, const float* __restrict__ u,
           float* __restrict__ comb) {
  int g = blockIdx.x, c = threadIdx.x;
  float inv = 1.0f / fmaxf(gcnt[g], 1.0f);
  comb[(long)g * DIM3 + c]           = gsum_e[(long)g * DIM + c] * inv;
  comb[(long)g * DIM3 + DIM + c]     = gsum_v[(long)g * DIM + c] * inv;
  comb[(long)g * DIM3 + 2 * DIM + c] = u[(long)g * DIM + c];
}

// ---------------------------------------------------------------------------
// 4) GEMM out = relu(A @ W^T + bias), fp32 via V_WMMA_F32_16X16X4_F32.
//    A: [M x K] row-major, W: [N x K] row-major (so B = W^T is K x N).
//    One wave (32 lanes) per 16x16 output tile; 8 waves per block.
//    A operand (ISA 7.12.2, 32-bit A 16x4): lanes 0-15 -> K = {k0,k0+1},
//    lanes 16-31 -> K = {k0+2,k0+3}, M = lane&15 striped across lanes.
//    B operand mirrors with N across lanes. C/D: VGPR r -> rows r, r+8.
// ---------------------------------------------------------------------------
__global__ void __launch_bounds__(256)
gemm_relu_wmma(const float* __restrict__ A, const float* __restrict__ W,
               const float* __restrict__ bias, float* __restrict__ out,
               int M, int N, int K) {
  int wave = threadIdx.x >> 5;
  int lane = threadIdx.x & 31;
  int tile = blockIdx.x * 8 + wave;
  int ntn  = N >> 4;
  int mt   = tile / ntn;
  int nt   = tile % ntn;
  if (mt >= (M >> 4)) return;          // uniform per wave; EXEC all-1s at WMMA

  int lhalf = lane >> 4;               // 0: K pair {0,1}, 1: K pair {2,3}
  int l15   = lane & 15;
  const float* Arow = A + (long)(mt * 16 + l15) * K + lhalf * 2;
  const float* Wrow = W + (long)(nt * 16 + l15) * K + lhalf * 2;

  v8f c = {0.f, 0.f, 0.f, 0.f, 0.f, 0.f, 0.f, 0.f};
  for (int k0 = 0; k0 < K; k0 += 4) {
    v2f a = *(const v2f*)(Arow + k0);
    v2f b = *(const v2f*)(Wrow + k0);
    // (neg_a, A, neg_b, B, c_mod, C, reuse_a, reuse_b)
    c = __builtin_amdgcn_wmma_f32_16x16x4_f32(false, a, false, b, (short)0, c,
                                              false, false);
  }

  int col = nt * 16 + l15;
  float bv = bias[col];
  int rowbase = mt * 16 + lhalf * 8;
  #pragma unroll
  for (int r = 0; r < 8; ++r) {
    float v = c[r] + bv;
    v = v > 0.f ? v : 0.f;
    out[(long)(rowbase + r) * N + col] = v;
  }
}

// ---------------------------------------------------------------------------
// 5) BatchNorm training-mode column stats: stats[col]=mean, stats[N+col]=rstd
// ---------------------------------------------------------------------------
__global__ void __launch_bounds__(256)
bn_stats(const float* __restrict__ h, float* __restrict__ stats, int M, int N) {
  __shared__ float ssum[256];
  __shared__ float ssq[256];
  int col = blockIdx.x;
  float s = 0.f, s2 = 0.f;
  for (int row = threadIdx.x; row < M; row += 256) {
    float v = h[(long)row * N + col];
    s += v;
    s2 += v * v;
  }
  ssum[threadIdx.x] = s;
  ssq[threadIdx.x]  = s2;
  __syncthreads();
  for (int off = 128; off > 0; off >>= 1) {
    if (threadIdx.x < off) {
      ssum[threadIdx.x] += ssum[threadIdx.x + off];
      ssq[threadIdx.x]  += ssq[threadIdx.x + off];
    }
    __syncthreads();
  }
  if (threadIdx.x == 0) {
    float mean = ssum[0] / (float)M;
    float var  = ssq[0] / (float)M - mean * mean;
    stats[col]     = mean;
    stats[N + col] = rsqrtf(var + 1e-5f);
  }
}

__global__ void __launch_bounds__(256)
bn_apply(const float* __restrict__ h, const float* __restrict__ stats,
         const float* __restrict__ gamma, const float* __restrict__ beta,
         float* __restrict__ out, int N) {
  int col = threadIdx.x;
  long i = (long)blockIdx.x * N + col;
  out[i] = (h[i] - stats[col]) * stats[N + col] * gamma[col] + beta[col];
}

// ---------------------------------------------------------------------------
extern "C" void kernel_launch(void* const* d_in, const int* in_sizes, int n_in,
                              void* d_out, int out_size, void* d_ws, size_t ws_size,
                              hipStream_t stream) {
  const float* x         = (const float*)d_in[0];
  const float* edge_attr = (const float*)d_in[1];
  const float* u         = (const float*)d_in[2];
  const float* W0        = (const float*)d_in[3];
  const float* b0        = (const float*)d_in[4];
  const float* W1        = (const float*)d_in[5];
  const float* b1        = (const float*)d_in[6];
  const float* W2        = (const float*)d_in[7];
  const float* b2        = (const float*)d_in[8];
  const float* g0        = (const float*)d_in[9];
  const float* be0       = (const float*)d_in[10];
  const float* g1        = (const float*)d_in[11];
  const float* be1       = (const float*)d_in[12];
  const float* g2        = (const float*)d_in[13];
  const float* be2       = (const float*)d_in[14];
  const int* edge_index  = (const int*)d_in[15];  // [2, E]; row 0 = src
  const int* batch       = (const int*)d_in[16];

  const int n_edges  = in_sizes[15] / 2;   // 1,600,000
  const int n_nodes  = in_sizes[16];       // 100,000
  const int n_graphs = in_sizes[2] / DIM;  // 2,048

  // Workspace layout (floats). Total ~29M floats (~116 MB).
  float* ws       = (float*)d_ws;
  float* node_sum = ws;                                   // n_nodes * DIM
  float* node_cnt = node_sum + (long)n_nodes * DIM;       // n_nodes
  float* gsum_e   = node_cnt + n_nodes;                   // n_graphs * DIM
  float* gsum_v   = gsum_e + (long)n_graphs * DIM;        // n_graphs * DIM
  float* gcnt     = gsum_v + (long)n_graphs * DIM;        // n_graphs
  float* comb     = gcnt + n_graphs;                      // n_graphs * 3*DIM
  float* bufP     = comb + (long)n_graphs * DIM3;         // n_graphs * DIM
  float* bufH     = bufP + (long)n_graphs * DIM;          // n_graphs * DIM
  float* stats    = bufH + (long)n_graphs * DIM;          // 2 * DIM

  // Zero all accumulators in one contiguous async memset (graph-capture safe).
  size_t zero_bytes =
      ((size_t)n_nodes * DIM + n_nodes + 2 * (size_t)n_graphs * DIM + n_graphs) *
      sizeof(float);
  hipMemsetAsync(ws, 0, zero_bytes, stream);

  edge_scatter<<<(n_edges + 3) / 4, 256, 0, stream>>>(
      edge_attr, edge_index, node_sum, node_cnt, n_edges);
  node_scatter<<<(n_nodes + 3) / 4, 256, 0, stream>>>(
      x, batch, node_sum, node_cnt, gsum_e, gsum_v, gcnt, n_nodes);
  comb_build<<<n_graphs, DIM, 0, stream>>>(gsum_e, gsum_v, gcnt, u, comb);

  const int tiles = (n_graphs / 16) * (DIM / 16);  // 2048 tiles
  const int gblk  = (tiles + 7) / 8;               // 8 waves/block

  // Layer 0: [G x 768] @ [768 x 256]
  gemm_relu_wmma<<<gblk, 256, 0, stream>>>(comb, W0, b0, bufP, n_graphs, DIM, DIM3);
  bn_stats<<<DIM, 256, 0, stream>>>(bufP, stats, n_graphs, DIM);
  bn_apply<<<n_graphs, DIM, 0, stream>>>(bufP, stats, g0, be0, bufH, DIM);

  // Layer 1
  gemm_relu_wmma<<<gblk, 256, 0, stream>>>(bufH, W1, b1, bufP, n_graphs, DIM, DIM);
  bn_stats<<<DIM, 256, 0, stream>>>(bufP, stats, n_graphs, DIM);
  bn_apply<<<n_graphs, DIM, 0, stream>>>(bufP, stats, g1, be1, bufH, DIM);

  // Layer 2 -> d_out
  gemm_relu_wmma<<<gblk, 256, 0, stream>>>(bufH, W2, b2, bufP, n_graphs, DIM, DIM);
  bn_stats<<<DIM, 256, 0, stream>>>(bufP, stats, n_graphs, DIM);
  bn_apply<<<n_graphs, DIM, 0, stream>>>(bufP, stats, g2, be2, (float*)d_out, DIM);
}